// PointCloudGenerationModule_40810779246885
// MI455X (gfx1250) — compile-verified
//
#include <hip/hip_runtime.h>
#include <hip/hip_bf16.h>
#include <math.h>

#define B_    4
#define S_    512
#define PH_   4096
#define NPTS  16384
#define D_    1024
#define G_    128
#define H_    8
#define L_    4
#define FF_   4096
#define GS_   32
#define HD_   128
#define M_    (B_ * G_)   // 512
#define TD_   (3 * D_)    // 3072

typedef float v2f __attribute__((ext_vector_type(2)));
typedef float v8f __attribute__((ext_vector_type(8)));

// ---------------------------------------------------------------------------
// 1) meanH[b][ph] = mean_s last_hidden[b][s][ph]   (linear commutes with mean)
// ---------------------------------------------------------------------------
__global__ void mean_s_kernel(const float* __restrict__ lh, float* __restrict__ meanH) {
    int idx = blockIdx.x * blockDim.x + threadIdx.x;   // b*PH + ph
    int b = idx / PH_;
    int ph = idx - b * PH_;
    const float* p = lh + (size_t)b * S_ * PH_ + ph;
    float s = 0.f;
    for (int t = 0; t < S_; ++t) s += p[(size_t)t * PH_];
    meanH[idx] = s * (1.0f / (float)S_);
}

// ---------------------------------------------------------------------------
// 2) agg[b][d] = dot(meanH[b], fp_w[d]) + fp_b[d]   (one wave per output)
// ---------------------------------------------------------------------------
__global__ void wavedot_kernel(const float* __restrict__ X, const float* __restrict__ W,
                               const float* __restrict__ bias, float* __restrict__ out,
                               int K, int Nout) {
    int lane = threadIdx.x & 31;
    int wave = threadIdx.x >> 5;
    int o = blockIdx.x * (blockDim.x >> 5) + wave;
    int b = o / Nout;
    int n = o - b * Nout;
    const float* x = X + (size_t)b * K;
    const float* w = W + (size_t)n * K;
    float acc = 0.f;
    for (int k = lane; k < K; k += 32) acc += x[k] * w[k];
    for (int off = 16; off > 0; off >>= 1) acc += __shfl_down(acc, off);
    if (lane == 0) out[o] = acc + bias[n];
}

// ---------------------------------------------------------------------------
// 3) Farthest point sampling: one workgroup (1024 thr) per batch.
//    Points + running dist live in registers (16/thread); argmax via LDS tree.
// ---------------------------------------------------------------------------
__global__ void __launch_bounds__(1024) fps_kernel(const float* __restrict__ xyz,
                                                   float* __restrict__ centers) {
    const int b   = blockIdx.x;
    const int tid = threadIdx.x;
    const int PPT = NPTS / 1024;                       // 16 points per thread
    const float* base = xyz + (size_t)b * NPTS * 3;

    float px[PPT], py[PPT], pz[PPT], dist[PPT];
    for (int i = 0; i < PPT; ++i) {
        int p = i * 1024 + tid;                        // strided -> coalesced
        px[i] = base[p * 3 + 0];
        py[i] = base[p * 3 + 1];
        pz[i] = base[p * 3 + 2];
        dist[i] = 1e10f;
    }

    __shared__ float s_v[1024];
    __shared__ int   s_i[1024];
    __shared__ float s_c[3];
    if (tid == 0) { s_c[0] = base[0]; s_c[1] = base[1]; s_c[2] = base[2]; }
    __syncthreads();

    for (int it = 0; it < G_; ++it) {
        float cx = s_c[0], cy = s_c[1], cz = s_c[2];
        if (tid == 0) {
            centers[(b * G_ + it) * 3 + 0] = cx;
            centers[(b * G_ + it) * 3 + 1] = cy;
            centers[(b * G_ + it) * 3 + 2] = cz;
        }
        float bv = -1.f;
        int   bi = NPTS;
        for (int i = 0; i < PPT; ++i) {
            float dx = px[i] - cx, dy = py[i] - cy, dz = pz[i] - cz;
            float d = dx * dx + dy * dy + dz * dz;
            float nd = fminf(dist[i], d);
            dist[i] = nd;
            int p = i * 1024 + tid;
            if (nd > bv) { bv = nd; bi = p; }
        }
        s_v[tid] = bv; s_i[tid] = bi;
        __syncthreads();
        for (int off = 512; off > 0; off >>= 1) {
            if (tid < off) {
                float v2 = s_v[tid + off]; int i2 = s_i[tid + off];
                if (v2 > s_v[tid] || (v2 == s_v[tid] && i2 < s_i[tid])) {
                    s_v[tid] = v2; s_i[tid] = i2;
                }
            }
            __syncthreads();
        }
        if (tid == 0) {
            int far = s_i[0];
            s_c[0] = base[far * 3 + 0];
            s_c[1] = base[far * 3 + 1];
            s_c[2] = base[far * 3 + 2];
        }
        __syncthreads();
    }
}

// ---------------------------------------------------------------------------
// LayerNorm over D=1024 (one block per row); optionally adds pos_embed[g].
// ---------------------------------------------------------------------------
__global__ void ln_kernel(const float* __restrict__ x, const float* __restrict__ pe,
                          const float* __restrict__ w, const float* __restrict__ b,
                          float* __restrict__ out) {
    int m = blockIdx.x, tid = threadIdx.x;
    int g = m % G_;
    __shared__ float r1[256], r2[256];
    const float* xr = x + (size_t)m * D_;
    const float* pr = pe ? (pe + (size_t)g * D_) : nullptr;
    float v[4];
    float s = 0.f, s2 = 0.f;
    for (int i = 0; i < 4; ++i) {
        int d = tid + i * 256;
        float t = xr[d] + (pr ? pr[d] : 0.f);
        v[i] = t; s += t; s2 += t * t;
    }
    r1[tid] = s; r2[tid] = s2; __syncthreads();
    for (int off = 128; off > 0; off >>= 1) {
        if (tid < off) { r1[tid] += r1[tid + off]; r2[tid] += r2[tid + off]; }
        __syncthreads();
    }
    float mean = r1[0] * (1.f / (float)D_);
    float var  = r2[0] * (1.f / (float)D_) - mean * mean;
    float rstd = rsqrtf(var + 1e-5f);
    for (int i = 0; i < 4; ++i) {
        int d = tid + i * 256;
        out[(size_t)m * D_ + d] = (v[i] - mean) * rstd * w[d] + b[d];
    }
}

// ---------------------------------------------------------------------------
// Generic WMMA GEMM:  C[m][n] = f( (sum_k A[m][k]*W[n][k]) * scale[n] + shift[n] )
//                               (+ residual[m][n])
// Wave tile: 16(M) x 64(N) via 4 fragments of V_WMMA_F32_16X16X4_F32.
// Block = 128 threads (4 waves) -> 16 x 256 tile.
// A-fragment layout (f32 16x4): lanes 0-15 carry K=0,1; lanes 16-31 carry K=2,3.
// OOB lanes use address-clamped (in-bounds) loads with NO value masking: an A
// lane only feeds its own C row and a B fragment only its own C column, so
// garbage stays in rows/cols the guarded epilogue never stores. EXEC remains
// all-ones through the WMMA loop (ISA requirement) and the loop body is pure
// loads + WMMA.
// nt != 0 -> stream W with non-temporal loads (single-use giant weights).
// ---------------------------------------------------------------------------
__global__ void __launch_bounds__(128) gemm_wmma_kernel(
    const float* __restrict__ A, const float* __restrict__ W,
    const float* __restrict__ shift, const float* __restrict__ scale,
    const float* __restrict__ residual, float* __restrict__ C,
    int M, int N, int K, int act, int nt) {
    const int lane  = threadIdx.x & 31;
    const int wave  = threadIdx.x >> 5;
    const int lm    = lane & 15;
    const int kh    = (lane >> 4) << 1;               // 0 or 2
    const int mbase = blockIdx.y * 16;
    const int nbase = blockIdx.x * 256 + wave * 64;

    const int  m   = mbase + lm;
    const bool mok = (m < M);
    const float* arow = A + (size_t)(mok ? m : 0) * K + kh;

    int  ncol[4]; bool nok[4];
    const float* wrow[4];
#pragma unroll
    for (int f = 0; f < 4; ++f) {
        ncol[f] = nbase + f * 16 + lm;
        nok[f]  = (ncol[f] < N);
        wrow[f] = W + (size_t)(nok[f] ? ncol[f] : 0) * K + kh;
    }

    const v8f vzero = {0.f, 0.f, 0.f, 0.f, 0.f, 0.f, 0.f, 0.f};
    v8f acc[4];
#pragma unroll
    for (int f = 0; f < 4; ++f) acc[f] = vzero;

    if (nt) {
        for (int k = 0; k < K; k += 4) {
            v2f a = *(const v2f*)(arow + k);
#pragma unroll
            for (int f = 0; f < 4; ++f) {
                v2f bb = __builtin_nontemporal_load((const v2f*)(wrow[f] + k));
                acc[f] = __builtin_amdgcn_wmma_f32_16x16x4_f32(
                    false, a, false, bb, (short)0, acc[f], false, false);
            }
        }
    } else {
        for (int k = 0; k < K; k += 4) {
            v2f a = *(const v2f*)(arow + k);
#pragma unroll
            for (int f = 0; f < 4; ++f) {
                v2f bb = *(const v2f*)(wrow[f] + k);
                acc[f] = __builtin_amdgcn_wmma_f32_16x16x4_f32(
                    false, a, false, bb, (short)0, acc[f], false, false);
            }
        }
    }

    // C/D layout: lane<16 -> rows mbase+0..7, lane>=16 -> rows mbase+8..15;
    // each lane owns one fixed column per fragment.
    const int rbase = mbase + ((lane >> 4) << 3);
#pragma unroll
    for (int f = 0; f < 4; ++f) {
        if (!nok[f]) continue;
        int n = ncol[f];
        float sc = scale ? scale[n] : 1.f;
        float sh = shift ? shift[n] : 0.f;
#pragma unroll
        for (int r = 0; r < 8; ++r) {
            int mm = rbase + r;
            if (mm >= M) continue;
            float vv = acc[f][r] * sc + sh;
            if (act == 1)      vv = 0.5f * vv * (1.f + erff(vv * 0.70710678118654752f));
            else if (act == 2) vv = fmaxf(vv, 0.f);
            if (residual) vv += residual[(size_t)mm * N + n];
            C[(size_t)mm * N + n] = vv;
        }
    }
}

// ---------------------------------------------------------------------------
// Attention: softmax(q.k^T * 1/sqrt(HD)) then @ v.  One block per (b,h,qrow).
// ---------------------------------------------------------------------------
__global__ void attn_scores_kernel(const float* __restrict__ qkv, float* __restrict__ P) {
    int idx = blockIdx.x;
    int qg = idx % G_;
    int h  = (idx / G_) % H_;
    int b  = idx / (G_ * H_);
    int tid = threadIdx.x;                            // key index 0..127
    __shared__ float qs[HD_];
    __shared__ float red[G_];
    qs[tid] = qkv[(size_t)(b * G_ + qg) * TD_ + h * HD_ + tid];
    __syncthreads();
    const float* kr = qkv + (size_t)(b * G_ + tid) * TD_ + D_ + h * HD_;
    float s = 0.f;
    for (int d = 0; d < HD_; ++d) s += qs[d] * kr[d];
    s *= 0.08838834764831845f;                        // 1/sqrt(128)
    red[tid] = s; __syncthreads();
    for (int off = 64; off > 0; off >>= 1) {
        if (tid < off) red[tid] = fmaxf(red[tid], red[tid + off]);
        __syncthreads();
    }
    float mx = red[0]; __syncthreads();
    float e = __expf(s - mx);
    red[tid] = e; __syncthreads();
    for (int off = 64; off > 0; off >>= 1) {
        if (tid < off) red[tid] += red[tid + off];
        __syncthreads();
    }
    P[(size_t)idx * G_ + tid] = e * (1.f / red[0]);
}

__global__ void attn_av_kernel(const float* __restrict__ qkv, const float* __restrict__ P,
                               float* __restrict__ o) {
    int idx = blockIdx.x;
    int qg = idx % G_;
    int h  = (idx / G_) % H_;
    int b  = idx / (G_ * H_);
    int tid = threadIdx.x;                            // d index 0..127
    __shared__ float as[G_];
    as[tid] = P[(size_t)idx * G_ + tid];
    __syncthreads();
    const float* vbase = qkv + (size_t)(b * G_) * TD_ + 2 * D_ + h * HD_ + tid;
    float acc = 0.f;
    for (int kg = 0; kg < G_; ++kg) acc += as[kg] * vbase[(size_t)kg * TD_];
    o[(size_t)(b * G_ + qg) * D_ + h * HD_ + tid] = acc;
}

// ---------------------------------------------------------------------------
// Fold conv bias + eval-mode BN into per-channel scale/shift for GEMM epilogue
// ---------------------------------------------------------------------------
__global__ void bnfold_kernel(const float* __restrict__ c1b, const float* __restrict__ g,
                              const float* __restrict__ beta, const float* __restrict__ mmean,
                              const float* __restrict__ var, float* __restrict__ scale,
                              float* __restrict__ shift) {
    int d = blockIdx.x * blockDim.x + threadIdx.x;
    if (d < D_) {
        float sc = g[d] * rsqrtf(var[d] + 1e-5f);
        scale[d] = sc;
        shift[d] = (c1b[d] - mmean[d]) * sc + beta[d];
    }
}

// ---------------------------------------------------------------------------
// Head: deltas = hp @ c2_w.T + c2_b (N=96), pred = deltas + centers broadcast.
// ---------------------------------------------------------------------------
__global__ void predict_kernel(const float* __restrict__ hp, const float* __restrict__ w,
                               const float* __restrict__ bias, const float* __restrict__ centers,
                               float* __restrict__ out) {
    int m = blockIdx.x;
    int n = threadIdx.x;                              // 0..95
    __shared__ float hs[D_];
    for (int i = n; i < D_; i += 96) hs[i] = hp[(size_t)m * D_ + i];
    __syncthreads();
    const float* wr = w + (size_t)n * D_;
    float acc = 0.f;
    for (int k = 0; k < D_; ++k) acc += hs[k] * wr[k];
    acc += bias[n];
    int sidx = n / 3, c = n - sidx * 3;
    acc += centers[m * 3 + c];
    out[((size_t)m * GS_ + sidx) * 3 + c] = acc;
}

// ---------------------------------------------------------------------------
extern "C" void kernel_launch(void* const* d_in, const int* in_sizes, int n_in,
                              void* d_out, int out_size, void* d_ws, size_t ws_size,
                              hipStream_t stream) {
    (void)in_sizes; (void)n_in; (void)out_size; (void)ws_size;
    const float* last_hidden = (const float*)d_in[0];
    const float* pc   = (const float*)d_in[1];
    const float* fp_w = (const float*)d_in[2];
    const float* fp_b = (const float*)d_in[3];
    const float* sp_w = (const float*)d_in[4];
    const float* sp_b = (const float*)d_in[5];
    const float* pos  = (const float*)d_in[6];
    const float* ln1w = (const float*)d_in[7];
    const float* ln1b = (const float*)d_in[8];
    const float* inw  = (const float*)d_in[9];
    const float* inb  = (const float*)d_in[10];
    const float* outw = (const float*)d_in[11];
    const float* outb = (const float*)d_in[12];
    const float* ln2w = (const float*)d_in[13];
    const float* ln2b = (const float*)d_in[14];
    const float* w1   = (const float*)d_in[15];
    const float* b1   = (const float*)d_in[16];
    const float* w2   = (const float*)d_in[17];
    const float* b2   = (const float*)d_in[18];
    const float* c1w  = (const float*)d_in[19];
    const float* c1b  = (const float*)d_in[20];
    const float* bng  = (const float*)d_in[21];
    const float* bnb  = (const float*)d_in[22];
    const float* bnm  = (const float*)d_in[23];
    const float* bnv  = (const float*)d_in[24];
    const float* c2w  = (const float*)d_in[25];
    const float* c2b  = (const float*)d_in[26];
    float* out = (float*)d_out;

    float* ws    = (float*)d_ws;
    float* meanH = ws; ws += B_ * PH_;
    float* agg   = ws; ws += B_ * D_;
    float* x     = ws; ws += (size_t)M_ * D_;
    float* xn    = ws; ws += (size_t)M_ * D_;
    float* qkv   = ws; ws += (size_t)M_ * TD_;
    float* P     = ws; ws += (size_t)B_ * H_ * G_ * G_;
    float* ao    = ws; ws += (size_t)M_ * D_;
    float* hbuf  = ws; ws += (size_t)M_ * FF_;
    float* cent  = ws; ws += B_ * G_ * 3;
    float* bnsc  = ws; ws += D_;
    float* bnsh  = ws; ws += D_;
    float* hp    = hbuf;                              // reuse (lifetimes disjoint)

    // 1) mean over S, 2) agg = meanH @ fp_w.T + fp_b
    mean_s_kernel<<<(B_ * PH_) / 256, 256, 0, stream>>>(last_hidden, meanH);
    wavedot_kernel<<<(B_ * D_) / 8, 256, 0, stream>>>(meanH, fp_w, fp_b, agg, PH_, D_);

    // 3) FPS centers
    fps_kernel<<<B_, 1024, 0, stream>>>(pc, cent);

    // 4) x = (agg @ sp_w.T + sp_b) viewed as [B*G, D]   (M=4, N=131072, K=1024)
    //    512 MB single-use weight -> non-temporal streaming.
    {
        dim3 g((G_ * D_ + 255) / 256, (B_ + 15) / 16);
        gemm_wmma_kernel<<<g, 128, 0, stream>>>(agg, sp_w, sp_b, nullptr, nullptr,
                                                x, B_, G_ * D_, D_, 0, 1);
    }

    // 5) transformer blocks
    for (int l = 0; l < L_; ++l) {
        ln_kernel<<<M_, 256, 0, stream>>>(x, pos, ln1w + l * D_, ln1b + l * D_, xn);
        {
            dim3 g((TD_ + 255) / 256, (M_ + 15) / 16);
            gemm_wmma_kernel<<<g, 128, 0, stream>>>(xn, inw + (size_t)l * TD_ * D_,
                                                    inb + l * TD_, nullptr, nullptr,
                                                    qkv, M_, TD_, D_, 0, 0);
        }
        attn_scores_kernel<<<B_ * H_ * G_, G_, 0, stream>>>(qkv, P);
        attn_av_kernel<<<B_ * H_ * G_, HD_, 0, stream>>>(qkv, P, ao);
        {
            dim3 g((D_ + 255) / 256, (M_ + 15) / 16);
            gemm_wmma_kernel<<<g, 128, 0, stream>>>(ao, outw + (size_t)l * D_ * D_,
                                                    outb + l * D_, nullptr, x,
                                                    x, M_, D_, D_, 0, 0);
        }
        ln_kernel<<<M_, 256, 0, stream>>>(x, nullptr, ln2w + l * D_, ln2b + l * D_, xn);
        {
            dim3 g((FF_ + 255) / 256, (M_ + 15) / 16);
            gemm_wmma_kernel<<<g, 128, 0, stream>>>(xn, w1 + (size_t)l * FF_ * D_,
                                                    b1 + l * FF_, nullptr, nullptr,
                                                    hbuf, M_, FF_, D_, 1, 0);
        }
        {
            dim3 g((D_ + 255) / 256, (M_ + 15) / 16);
            gemm_wmma_kernel<<<g, 128, 0, stream>>>(hbuf, w2 + (size_t)l * D_ * FF_,
                                                    b2 + l * D_, nullptr, x,
                                                    x, M_, D_, FF_, 0, 0);
        }
    }

    // 6) predictor head: BN-folded conv1 (+ReLU) then conv2 + centers
    bnfold_kernel<<<(D_ + 255) / 256, 256, 0, stream>>>(c1b, bng, bnb, bnm, bnv, bnsc, bnsh);
    {
        dim3 g((D_ + 255) / 256, (M_ + 15) / 16);
        gemm_wmma_kernel<<<g, 128, 0, stream>>>(x, c1w, bnsh, bnsc, nullptr,
                                                hp, M_, D_, D_, 2, 0);
    }
    predict_kernel<<<M_, 3 * GS_, 0, stream>>>(hp, c2w, c2b, cent, out);
}